// decoder_21990232556015
// MI455X (gfx1250) — compile-verified
//
#include <hip/hip_runtime.h>
#include <hip/hip_bf16.h>

// Problem constants (from the reference)
#define PB    16      // batch
#define PK    8       // slots per batch
#define PN    196     // patches
#define PD    768     // token dim
#define PH    3072    // hidden dim
#define PDOUT 768     // output dim

// GEMM tiling
#define BM   64       // block tile M
#define BN   128      // block tile N
#define BKC  32       // K chunk staged in LDS
#define TSTR 36       // LDS row stride (32 + 4 pad; 36*4=144B, 16B aligned)

typedef __attribute__((ext_vector_type(2))) float v2f;
typedef __attribute__((ext_vector_type(8))) float v8f;

// ---- CDNA5 async global->LDS copy (ASYNCcnt-tracked, no VGPR round trip) --
// Flat pointers to __shared__ carry the DS byte address in their low 32 bits
// (ISA 10.2: LDS aperture base occupies addr[63:32]).
__device__ __forceinline__ void async_copy_b128(const float* g, float* l) {
    asm volatile("global_load_async_to_lds_b128 %0, %1, off"
                 :: "v"((unsigned)(unsigned long long)l),
                    "v"((unsigned long long)g)
                 : "memory");
}

__device__ __forceinline__ void wait_async0() {
#if __has_builtin(__builtin_amdgcn_s_wait_asynccnt)
    __builtin_amdgcn_s_wait_asynccnt(0);
#else
    asm volatile("s_wait_asynccnt 0" ::: "memory");
#endif
}

// ---------------------------------------------------------------------------
// fp32 WMMA GEMM with pre-transposed B:
//   C[M x N] = scale * (A[M x K] @ B[K x N]) + bias,  B given as Bt[N x K].
//   256 threads = 8 waves (2 x 4); wave tile 32(M) x 32(N) = four 16x16 accs.
//   Both A and Bt staged row-major in LDS -> every fragment is ONE aligned
//   ds_load_b64 (A: As[m][k..k+1], B: Bs[n][k..k+1]); 4 DS b64 per 4 WMMAs.
//   K staged via double-buffered async-to-LDS chunks of 32.
// ---------------------------------------------------------------------------
__global__ __launch_bounds__(256)
void gemm_f32_wmma(const float* __restrict__ A, int lda,
                   const float* __restrict__ Bt, int ldbt,   // ldbt == Kdim
                   float* __restrict__ C, int ldc,
                   int M, int Kdim,
                   const float* __restrict__ bias, float scale) {
    __shared__ float As[2][BM * TSTR];    // 2 x  9 KB
    __shared__ float Bs[2][BN * TSTR];    // 2 x 18 KB   (total 54 KB)

    const int tid  = threadIdx.x;
    const int lane = tid & 31;
    const int w    = tid >> 5;            // wave 0..7
    const int mt   = w >> 2;              // 0..1 : 32-row M subtile
    const int nt   = w & 3;               // 0..3 : 32-col N subtile

    const int n0 = blockIdx.x * BN;
    const int m0 = blockIdx.y * BM;

    const int lrow  = lane & 15;
    const int lhalf = lane >> 4;

    const int Mclamp = M - 1;

    auto prefetch = [&](int k0, int buf) {
        // A chunk: BM(64) x BKC(32) = 512 float4 -> 2 per thread
        #pragma unroll
        for (int i = 0; i < 2; ++i) {
            int idx = tid + i * 256;
            int row = idx >> 3;                       // 0..63
            int c4  = (idx & 7) << 2;                 // 0..28
            int gr  = m0 + row; if (gr > Mclamp) gr = Mclamp;
            async_copy_b128(&A[(size_t)gr * lda + k0 + c4],
                            &As[buf][row * TSTR + c4]);
        }
        // Bt chunk: BN(128) x BKC(32) = 1024 float4 -> 4 per thread
        #pragma unroll
        for (int i = 0; i < 4; ++i) {
            int idx = tid + i * 256;
            int row = idx >> 3;                       // 0..127 (n index)
            int c4  = (idx & 7) << 2;
            async_copy_b128(&Bt[(size_t)(n0 + row) * ldbt + k0 + c4],
                            &Bs[buf][row * TSTR + c4]);
        }
    };

    v8f acc00 = {}, acc01 = {}, acc10 = {}, acc11 = {};

    const int am0 = mt * 32 + lrow;                   // A rows
    const int am1 = am0 + 16;
    const int bn0 = nt * 32 + lrow;                   // B cols
    const int bn1 = bn0 + 16;

    const int nchunks = Kdim / BKC;
    prefetch(0, 0);

    for (int c = 0; c < nchunks; ++c) {
        const int buf = c & 1;
        wait_async0();
        __syncthreads();                              // buf data visible to all
        if (c + 1 < nchunks) prefetch((c + 1) * BKC, buf ^ 1);

        const float* as = &As[buf][0];
        const float* bs = &Bs[buf][0];
        #pragma unroll
        for (int kk = 0; kk < BKC; kk += 4) {
            const int ko = kk + lhalf * 2;            // VGPR0: K=ko, VGPR1: K=ko+1
            v2f a0 = *reinterpret_cast<const v2f*>(&as[am0 * TSTR + ko]);
            v2f a1 = *reinterpret_cast<const v2f*>(&as[am1 * TSTR + ko]);
            v2f b0 = *reinterpret_cast<const v2f*>(&bs[bn0 * TSTR + ko]);
            v2f b1 = *reinterpret_cast<const v2f*>(&bs[bn1 * TSTR + ko]);
            acc00 = __builtin_amdgcn_wmma_f32_16x16x4_f32(
                        false, a0, false, b0, (short)0, acc00, false, false);
            acc01 = __builtin_amdgcn_wmma_f32_16x16x4_f32(
                        false, a0, false, b1, (short)0, acc01, false, false);
            acc10 = __builtin_amdgcn_wmma_f32_16x16x4_f32(
                        false, a1, false, b0, (short)0, acc10, false, false);
            acc11 = __builtin_amdgcn_wmma_f32_16x16x4_f32(
                        false, a1, false, b1, (short)0, acc11, false, false);
        }
        __syncthreads();                              // all reads of buf done
    }

    // ---- epilogue: C layout VGPR r -> M = r + 8*(L/16), N = L%16 ---------
    const int gcol0 = n0 + nt * 32 + lrow;
    const int gcol1 = gcol0 + 16;
    const float bv0 = bias ? bias[gcol0] : 0.0f;
    const float bv1 = bias ? bias[gcol1] : 0.0f;
    #pragma unroll
    for (int r = 0; r < 8; ++r) {
        const int grow0 = m0 + mt * 32 + r + lhalf * 8;
        const int grow1 = grow0 + 16;
        if (grow0 < M) {
            C[(size_t)grow0 * ldc + gcol0] = acc00[r] * scale + bv0;
            C[(size_t)grow0 * ldc + gcol1] = acc01[r] * scale + bv1;
        }
        if (grow1 < M) {
            C[(size_t)grow1 * ldc + gcol0] = acc10[r] * scale + bv0;
            C[(size_t)grow1 * ldc + gcol1] = acc11[r] * scale + bv1;
        }
    }
}

// ---------------------------------------------------------------------------
// 32x32 LDS-tiled transpose: out[c][r] = in[r][c], rows/cols multiples of 32.
// ---------------------------------------------------------------------------
__global__ __launch_bounds__(256)
void transpose32(const float* __restrict__ in, int rows, int cols,
                 float* __restrict__ out) {
    __shared__ float t[32][33];
    const int c0 = blockIdx.x * 32;
    const int r0 = blockIdx.y * 32;
    const int x = threadIdx.x & 31;
    const int y = threadIdx.x >> 5;                   // 0..7
    #pragma unroll
    for (int i = 0; i < 4; ++i)
        t[y + 8 * i][x] = in[(size_t)(r0 + y + 8 * i) * cols + c0 + x];
    __syncthreads();
    #pragma unroll
    for (int i = 0; i < 4; ++i)
        out[(size_t)(c0 + y + 8 * i) * rows + r0 + x] = t[x][y + 8 * i];
}

// ---------------------------------------------------------------------------
// hbar[b*196+n, h] = sum_{k<8} relu(S1[b*8+k, h] + P1[n, h])
// (b1 already folded into S1). float4 per thread.
// ---------------------------------------------------------------------------
__global__ __launch_bounds__(256)
void hbar_reduce(const float* __restrict__ S1,   // 128 x 3072  (slots@W1 + b1)
                 const float* __restrict__ P1,   // 196 x 3072  (pos@W1)
                 float* __restrict__ H) {        // 3136 x 3072
    const int idx = blockIdx.x * 256 + threadIdx.x;   // over 3136*768 float4s
    const int h4  = idx % (PH / 4);
    const int row = idx / (PH / 4);                   // bn = b*196 + n
    const int b   = row / PN;
    const int n   = row - b * PN;
    const int h   = h4 * 4;

    const float4 p = *reinterpret_cast<const float4*>(&P1[(size_t)n * PH + h]);
    float4 acc = make_float4(0.f, 0.f, 0.f, 0.f);
    #pragma unroll
    for (int k = 0; k < PK; ++k) {
        const float4 s = *reinterpret_cast<const float4*>(&S1[(size_t)(b * PK + k) * PH + h]);
        acc.x += fmaxf(s.x + p.x, 0.f);
        acc.y += fmaxf(s.y + p.y, 0.f);
        acc.z += fmaxf(s.z + p.z, 0.f);
        acc.w += fmaxf(s.w + p.w, 0.f);
    }
    *reinterpret_cast<float4*>(&H[(size_t)row * PH + h]) = acc;
}

// ---------------------------------------------------------------------------
extern "C" void kernel_launch(void* const* d_in, const int* in_sizes, int n_in,
                              void* d_out, int out_size, void* d_ws, size_t ws_size,
                              hipStream_t stream) {
    const float* slots = (const float*)d_in[0];   // (16,8,768)  -> 128 x 768
    const float* pos   = (const float*)d_in[1];   // (196,768)
    // d_in[2] = map_alpha : softmax over K of K-independent tensor == 1/K; unused
    const float* W1    = (const float*)d_in[3];   // (768,3072)
    const float* b1    = (const float*)d_in[4];   // (3072)
    const float* W2    = (const float*)d_in[5];   // (3072,768)
    const float* b2    = (const float*)d_in[6];   // (768)
    float* out = (float*)d_out;                   // (16,196,768) -> 3136 x 768

    float* S1  = (float*)d_ws;                    // 128  x 3072
    float* P1  = S1  + (size_t)PB * PK * PH;      // 196  x 3072
    float* HB  = P1  + (size_t)PN * PH;           // 3136 x 3072
    float* W1t = HB  + (size_t)PB * PN * PH;      // 3072 x 768
    float* W2t = W1t + (size_t)PD * PH;           // 768  x 3072

    // Pre-transpose weights so GEMM B-tiles stage as [n][k] with b128 asyncs.
    transpose32<<<dim3(PH / 32, PD / 32), 256, 0, stream>>>(W1, PD, PH, W1t);
    transpose32<<<dim3(PDOUT / 32, PH / 32), 256, 0, stream>>>(W2, PH, PDOUT, W2t);

    // S1 = slots @ W1 + b1   (128 x 3072)
    gemm_f32_wmma<<<dim3(PH / BN, (PB * PK) / BM), 256, 0, stream>>>(
        slots, PD, W1t, PD, S1, PH, PB * PK, PD, b1, 1.0f);

    // P1 = pos_embed @ W1    (196 x 3072), ragged M -> 4 row-tiles of 64
    gemm_f32_wmma<<<dim3(PH / BN, (PN + BM - 1) / BM), 256, 0, stream>>>(
        pos, PD, W1t, PD, P1, PH, PN, PD, nullptr, 1.0f);

    // hbar = sum_k relu(S1 + P1)   (3136 x 3072)
    hbar_reduce<<<(PB * PN * (PH / 4)) / 256, 256, 0, stream>>>(S1, P1, HB);

    // y = (1/8) * hbar @ W2 + b2   (3136 x 768)
    gemm_f32_wmma<<<dim3(PDOUT / BN, (PB * PN) / BM), 256, 0, stream>>>(
        HB, PH, W2t, PH, out, PDOUT, PB * PN, PH, b2, 1.0f / PK);
}